// ContrastiveLoss_67413806678513
// MI455X (gfx1250) — compile-verified
//
#include <hip/hip_runtime.h>

typedef __attribute__((ext_vector_type(2))) float v2f;
typedef __attribute__((ext_vector_type(4))) float v4f;
typedef __attribute__((ext_vector_type(8))) float v8f;

#define BATCH 16
#define NDIM  1024
#define DDIM  512

// ---------------------------------------------------------------------------
// Kernel 1: per-row squared L2 norms of a and b -> workspace.
// One wave per 512-float row; float4 loads; shuffle reduction.
// Block 0 / thread 0 also zeroes the output accumulator.
// ---------------------------------------------------------------------------
__global__ void __launch_bounds__(256)
row_norms_kernel(const float* __restrict__ a, const float* __restrict__ b,
                 float* __restrict__ a2, float* __restrict__ b2,
                 float* __restrict__ out) {
    const int wave = threadIdx.x >> 5;
    const int lane = threadIdx.x & 31;
    const int row  = blockIdx.x * 8 + wave;          // 0 .. 2*16384-1

    const float* src;
    float* dst;
    if (row < BATCH * NDIM) {
        src = a + (size_t)row * DDIM;
        dst = a2 + row;
    } else {
        const int r = row - BATCH * NDIM;
        src = b + (size_t)r * DDIM;
        dst = b2 + r;
    }

    const v4f* p = (const v4f*)src;                  // 128 float4 per row
    float s = 0.0f;
#pragma unroll
    for (int j = 0; j < 4; ++j) {
        v4f v = p[lane + 32 * j];
        s += v.x * v.x + v.y * v.y + v.z * v.z + v.w * v.w;
    }
#pragma unroll
    for (int off = 16; off > 0; off >>= 1)
        s += __shfl_down(s, off, 32);
    if (lane == 0) *dst = s;

    if (blockIdx.x == 0 && threadIdx.x == 0) *out = 0.0f;
}

// ---------------------------------------------------------------------------
// Kernel 2: fused batched GEMM (a @ b^T via V_WMMA_F32_16X16X4_F32) +
// contrastive-loss epilogue + scalar reduction.
//
// Grid: (32 tiles of 128x256, 16 batches). Block: 256 threads = 8 waves,
// arranged 2(M) x 4(N); each wave computes a 64x64 output region as a
// 4x4 grid of 16x16 WMMA accumulators (never written to memory).
// Per K-step: 8 float2 fragment loads feed 16 WMMAs -> 16 FLOP per L2 byte.
//
// WMMA f32 16x16x4 operand layout (ISA 7.12.2):
//   A 16x4 : lane l holds A[l%16][kbase + 2*(l/16)] in v0, ...+1 in v1
//            -> contiguous float2 load from a's row.
//   B 4x16 : lane l holds B[kbase + 2*(l/16)][l%16]; our B is b^T, so this
//            is b[col][k] -> contiguous float2 load from b's row.
//   C/D    : VGPR i, lanes 0-15 -> (M=i, N=lane); lanes 16-31 -> (M=i+8).
// ---------------------------------------------------------------------------
__global__ void __launch_bounds__(256)
contrastive_gemm_kernel(const float* __restrict__ a, const float* __restrict__ b,
                        const float* __restrict__ a2, const float* __restrict__ b2,
                        float* __restrict__ out) {
    const int batch = blockIdx.y;
    const int tileM = (blockIdx.x & 7) << 7;         // 8 M-tiles of 128
    const int tileN = (blockIdx.x >> 3) << 8;        // 4 N-tiles of 256

    const int wave = threadIdx.x >> 5;
    const int lane = threadIdx.x & 31;
    const int lr   = lane & 15;                      // row/col within 16
    const int lg   = lane >> 4;                      // k-group: 0 or 1

    const int m0 = tileM + (wave & 1) * 64;          // wave's 64-row strip
    const int n0 = tileN + (wave >> 1) * 64;         // wave's 64-col strip

    const float* A  = a + (size_t)batch * NDIM * DDIM;
    const float* Bm = b + (size_t)batch * NDIM * DDIM;

    // Per-lane base pointers for fragment loads (float2 at k + 2*lg).
    const float* ap[4];
    const float* bp[4];
#pragma unroll
    for (int t = 0; t < 4; ++t) {
        ap[t] = A  + (size_t)(m0 + t * 16 + lr) * DDIM + 2 * lg;
        bp[t] = Bm + (size_t)(n0 + t * 16 + lr) * DDIM + 2 * lg;
    }

    v8f acc[4][4];
#pragma unroll
    for (int i = 0; i < 4; ++i)
#pragma unroll
        for (int j = 0; j < 4; ++j)
            acc[i][j] = (v8f)(0.0f);

#pragma unroll 2
    for (int kk = 0; kk < DDIM; kk += 4) {
        v2f af[4], bf[4];
#pragma unroll
        for (int t = 0; t < 4; ++t) {
            af[t] = *(const v2f*)(ap[t] + kk);
            bf[t] = *(const v2f*)(bp[t] + kk);
        }
#pragma unroll
        for (int i = 0; i < 4; ++i)
#pragma unroll
            for (int j = 0; j < 4; ++j)
                acc[i][j] = __builtin_amdgcn_wmma_f32_16x16x4_f32(
                    false, af[i], false, bf[j], (short)0, acc[i][j], false, false);
    }

    // Epilogue: d2 = a2[m] + b2[n] - 2*ab ; loss terms for m != n.
    const float* a2b = a2 + batch * NDIM;
    const float* b2b = b2 + batch * NDIM;

    float am[4][8];
#pragma unroll
    for (int i = 0; i < 4; ++i) {
        const int mbase = m0 + i * 16 + lg * 8;
#pragma unroll
        for (int e = 0; e < 8; ++e) am[i][e] = a2b[mbase + e];
    }

    float partial = 0.0f;
#pragma unroll
    for (int j = 0; j < 4; ++j) {
        const int n  = n0 + j * 16 + lr;
        const float bn = b2b[n];
#pragma unroll
        for (int i = 0; i < 4; ++i) {
            const int mbase = m0 + i * 16 + lg * 8;
#pragma unroll
            for (int e = 0; e < 8; ++e) {
                const int m = mbase + e;
                float d2 = am[i][e] + bn - 2.0f * acc[i][j][e];
                d2 = fmaxf(d2, 0.0f);
                const float d = sqrtf(d2);
                const float h = fmaxf(1.0f - d, 0.0f);   // MARGIN = 1.0
                const float t = d2 + h * h;
                partial += (m != n) ? t : 0.0f;
            }
        }
    }

    const float inv_nneg = 1.0f / ((float)BATCH * (float)NDIM * (float)(NDIM - 1));
    partial *= inv_nneg;

#pragma unroll
    for (int off = 16; off > 0; off >>= 1)
        partial += __shfl_down(partial, off, 32);
    if (lane == 0) atomicAdd(out, partial);
}

// ---------------------------------------------------------------------------
// Launch: kernel1 (norms + zero out) then kernel2 (fused GEMM + loss).
// Workspace use: 2 * 16384 floats = 128 KB.
// ---------------------------------------------------------------------------
extern "C" void kernel_launch(void* const* d_in, const int* in_sizes, int n_in,
                              void* d_out, int out_size, void* d_ws, size_t ws_size,
                              hipStream_t stream) {
    const float* a = (const float*)d_in[0];
    const float* b = (const float*)d_in[1];
    float* out = (float*)d_out;
    float* ws  = (float*)d_ws;
    float* a2  = ws;
    float* b2  = ws + BATCH * NDIM;

    // 2 * 16384 rows, 8 rows (waves) per block
    row_norms_kernel<<<(2 * BATCH * NDIM) / 8, 256, 0, stream>>>(a, b, a2, b2, out);

    dim3 grid(32, BATCH);   // 8(M) x 4(N) tiles of 128x256 per batch
    contrastive_gemm_kernel<<<grid, 256, 0, stream>>>(a, b, a2, b2, out);
}